// GMMNet_29188597744213
// MI455X (gfx1250) — compile-verified
//
#include <hip/hip_runtime.h>
#include <hip/hip_bf16.h>
#include <math.h>

typedef float v2f __attribute__((ext_vector_type(2)));
typedef float v8f __attribute__((ext_vector_type(8)));

#define KMIX 9
#define HID  16
#define EPSF 1e-15f
#define EPT  4          // edges per thread in edge1

// ---------------------------------------------------------------------------
// Layer-1 edge kernel: gaussian weights + WMMA message GEMM + scatter atomics
//   msg[e, o] = sum_j u[e,j] * Gf[j,o],  u[e, i*9+k] = x[row[e],i] * w[e,k]
//   Gf = g1 reshaped [18,16] (padded to K=20 for 5x V_WMMA_F32_16X16X4_F32)
// ---------------------------------------------------------------------------
__global__ __launch_bounds__(256) void edge1_kernel(
    const float* __restrict__ x,
    const int*   __restrict__ row,
    const int*   __restrict__ col,
    const float* __restrict__ g1,     // [2,9,16] == Gf[18][16]
    const float* __restrict__ mu1,    // [9,2]
    const float* __restrict__ sigma1, // [9,2]
    float* __restrict__ agg1,         // [N,16], zero-initialized
    float* __restrict__ deg,          // [N], zero-initialized
    int E)
{
    __shared__ float ldsU[8][32][20];  // stride-20 rows: conflict-free on 64 banks
    __shared__ int4  ldsC[8][8];       // 32 dst-node ids per wave, int4-packed

    const int tid  = threadIdx.x;
    const int wv   = tid >> 5;
    const int lane = tid & 31;
    const int m    = lane & 15;   // A row (edge-in-tile) / B,C col (o)
    const int hh   = lane >> 4;   // lane-half: K offset (A,B) / M offset (C)

    const float2* __restrict__ x2  = (const float2*)x;
    const float2* __restrict__ mu2 = (const float2*)mu1;
    const float2* __restrict__ sg2 = (const float2*)sigma1;

    // ---- hoisted per-thread constants --------------------------------------
    // B operand slices: Bs[s][v] = Gf[4s + v + 2*hh][m], zero-padded rows 18,19
    v2f Bs[5];
#pragma unroll
    for (int s = 0; s < 5; ++s) {
#pragma unroll
        for (int v = 0; v < 2; ++v) {
            const int j = 4 * s + v + 2 * hh;
            Bs[s][v] = (j < 18) ? g1[j * HID + m] : 0.0f;
        }
    }
    float gm0[KMIX], gm1[KMIX], gi0[KMIX], gi1[KMIX];
#pragma unroll
    for (int k = 0; k < KMIX; ++k) {
        const float2 mu = mu2[k];
        const float2 sg = sg2[k];
        gm0[k] = mu.x;  gm1[k] = mu.y;
        gi0[k] = -0.5f / (EPSF + sg.x * sg.x);
        gi1[k] = -0.5f / (EPSF + sg.y * sg.y);
    }

    const int ebase = blockIdx.x * (256 * EPT);

    for (int it = 0; it < EPT; ++it) {          // uniform trip count
        const int e = ebase + it * 256 + tid;
        const bool valid = (e < E);

        int c = 0;
        float xr0 = 0.0f, xr1 = 0.0f, p0 = 0.0f, p1 = 0.0f;
        if (valid) {
            const int r = row[e];
            c = col[e];
            const float2 xr = x2[r];
            const float2 xc = x2[c];
            xr0 = xr.x; xr1 = xr.y;
            p0 = xr.x - xc.x;
            p1 = xr.y - xc.y;
        }

        float u[20];
#pragma unroll
        for (int k = 0; k < KMIX; ++k) {
            const float d0 = p0 - gm0[k];
            const float d1 = p1 - gm1[k];
            const float wk = __expf(gi0[k] * d0 * d0 + gi1[k] * d1 * d1);
            u[k]        = xr0 * wk;   // == 0 for invalid lanes (xr0 = 0)
            u[KMIX + k] = xr1 * wk;
        }
        u[18] = 0.0f; u[19] = 0.0f;

#pragma unroll
        for (int j = 0; j < 20; ++j) ldsU[wv][lane][j] = u[j];
        ((int*)&ldsC[wv][0])[lane] = c;
        __syncthreads();

        // Two 16-edge tiles per wave, 5x WMMA f32 16x16x4 each
        v8f acc[2];
        int cols[2][8];
#pragma unroll
        for (int t = 0; t < 2; ++t) {
            v8f a8 = {};
#pragma unroll
            for (int s = 0; s < 5; ++s) {
                v2f a;
                a[0] = ldsU[wv][t * 16 + m][4 * s + 0 + 2 * hh];
                a[1] = ldsU[wv][t * 16 + m][4 * s + 1 + 2 * hh];
                a8 = __builtin_amdgcn_wmma_f32_16x16x4_f32(
                    false, a, false, Bs[s], (short)0, a8, false, false);
            }
            acc[t] = a8;
            // 8 consecutive col ids this lane scatters: el = t*16 + hh*8 + jj
            const int4 ca = ldsC[wv][t * 4 + hh * 2 + 0];
            const int4 cb = ldsC[wv][t * 4 + hh * 2 + 1];
            cols[t][0] = ca.x; cols[t][1] = ca.y; cols[t][2] = ca.z; cols[t][3] = ca.w;
            cols[t][4] = cb.x; cols[t][5] = cb.y; cols[t][6] = cb.z; cols[t][7] = cb.w;
        }
        __syncthreads();   // LDS reuse safe for next iteration

        // Branch-free scatter: invalid edges carry value 0.0f into agg1[0..15]
#pragma unroll
        for (int t = 0; t < 2; ++t) {
#pragma unroll
            for (int jj = 0; jj < 8; ++jj) {
                atomicAdd(&agg1[cols[t][jj] * HID + m], acc[t][jj]);
            }
        }
        if (valid) atomicAdd(&deg[c], 1.0f);
    }
}

// ---------------------------------------------------------------------------
// Node kernel: mean-aggregate, root/bias, ELU; fold layer-2 node-side terms:
//   q[n,k] = sum_o h[n,o] * g2[o,k],  s[n] = h[n,:] . root2
// ---------------------------------------------------------------------------
__global__ __launch_bounds__(256) void node1_kernel(
    const float* __restrict__ x,
    const float* __restrict__ agg1,
    const float* __restrict__ deg,
    const float* __restrict__ root1,  // [2,16]
    const float* __restrict__ bias1,  // [16]
    const float* __restrict__ g2,     // [16,9,1]
    const float* __restrict__ root2,  // [16,1]
    float* __restrict__ q,            // [N,9]
    float* __restrict__ s,            // [N]
    int N)
{
    const int n = blockIdx.x * blockDim.x + threadIdx.x;
    if (n >= N) return;
    const float invd = 1.0f / fmaxf(deg[n], 1.0f);
    const float2 xv = ((const float2*)x)[n];

    float qk[KMIX];
#pragma unroll
    for (int k = 0; k < KMIX; ++k) qk[k] = 0.0f;
    float sv = 0.0f;

#pragma unroll
    for (int o = 0; o < HID; ++o) {
        const float v = agg1[n * HID + o] * invd
                      + xv.x * root1[o] + xv.y * root1[HID + o] + bias1[o];
        const float h = (v > 0.0f) ? v : (__expf(v) - 1.0f);  // ELU(alpha=1)
        sv += h * root2[o];
#pragma unroll
        for (int k = 0; k < KMIX; ++k) qk[k] += h * g2[o * KMIX + k];
    }
#pragma unroll
    for (int k = 0; k < KMIX; ++k) q[n * KMIX + k] = qk[k];
    s[n] = sv;
}

// ---------------------------------------------------------------------------
// Layer-2 edge kernel: msg = w2 . q[row], one atomic per edge
// ---------------------------------------------------------------------------
__global__ __launch_bounds__(256) void edge2_kernel(
    const float* __restrict__ x,
    const int*   __restrict__ row,
    const int*   __restrict__ col,
    const float* __restrict__ mu2,
    const float* __restrict__ sigma2,
    const float* __restrict__ q,
    float* __restrict__ agg2,
    int E)
{
    const int e = blockIdx.x * blockDim.x + threadIdx.x;
    if (e >= E) return;
    const int r = row[e], c = col[e];
    const float2 xr = ((const float2*)x)[r];
    const float2 xc = ((const float2*)x)[c];
    const float p0 = xr.x - xc.x;
    const float p1 = xr.y - xc.y;

    float msg = 0.0f;
#pragma unroll
    for (int k = 0; k < KMIX; ++k) {
        const float2 mu = ((const float2*)mu2)[k];
        const float2 sg = ((const float2*)sigma2)[k];
        const float d0 = p0 - mu.x;
        const float d1 = p1 - mu.y;
        const float i0 = -0.5f / (EPSF + sg.x * sg.x);
        const float i1 = -0.5f / (EPSF + sg.y * sg.y);
        msg += __expf(i0 * d0 * d0 + i1 * d1 * d1) * q[r * KMIX + k];
    }
    atomicAdd(&agg2[c], msg);
}

__global__ __launch_bounds__(256) void out_kernel(
    const float* __restrict__ agg2,
    const float* __restrict__ deg,
    const float* __restrict__ s,
    const float* __restrict__ bias2,
    float* __restrict__ out,
    int N)
{
    const int n = blockIdx.x * blockDim.x + threadIdx.x;
    if (n >= N) return;
    out[n] = agg2[n] / fmaxf(deg[n], 1.0f) + s[n] + bias2[0];
}

// ---------------------------------------------------------------------------
extern "C" void kernel_launch(void* const* d_in, const int* in_sizes, int n_in,
                              void* d_out, int out_size, void* d_ws, size_t ws_size,
                              hipStream_t stream) {
    const float* x      = (const float*)d_in[0];
    const int*   ei     = (const int*)  d_in[1];
    const float* g1     = (const float*)d_in[2];
    const float* mu1    = (const float*)d_in[3];
    const float* sigma1 = (const float*)d_in[4];
    const float* root1  = (const float*)d_in[5];
    const float* bias1  = (const float*)d_in[6];
    const float* g2     = (const float*)d_in[7];
    const float* mu2    = (const float*)d_in[8];
    const float* sigma2 = (const float*)d_in[9];
    const float* root2  = (const float*)d_in[10];
    const float* bias2  = (const float*)d_in[11];

    const int N = in_sizes[0] / 2;   // x: [N,2]
    const int E = in_sizes[1] / 2;   // edge_index: [2,E]
    const int* row = ei;
    const int* col = ei + E;

    // workspace layout (floats): deg[N] | agg1[16N] | agg2[N] | q[9N] | s[N]
    float* ws   = (float*)d_ws;
    float* deg  = ws;
    float* agg1 = ws + (size_t)N;
    float* agg2 = ws + (size_t)17 * N;
    float* q    = ws + (size_t)18 * N;
    float* s    = ws + (size_t)27 * N;

    // zero deg + agg1 + agg2 (contiguous first 18N floats)
    hipMemsetAsync(ws, 0, (size_t)18 * N * sizeof(float), stream);

    const int TB = 256;
    const int edges_per_block = TB * EPT;
    edge1_kernel<<<(E + edges_per_block - 1) / edges_per_block, TB, 0, stream>>>(
        x, row, col, g1, mu1, sigma1, agg1, deg, E);
    node1_kernel<<<(N + TB - 1) / TB, TB, 0, stream>>>(
        x, agg1, deg, root1, bias1, g2, root2, q, s, N);
    edge2_kernel<<<(E + TB - 1) / TB, TB, 0, stream>>>(
        x, row, col, mu2, sigma2, q, agg2, E);
    out_kernel<<<(N + TB - 1) / TB, TB, 0, stream>>>(
        agg2, deg, s, bias2, (float*)d_out, N);
}